// Projector_18064632447531
// MI455X (gfx1250) — compile-verified
//
#include <hip/hip_runtime.h>
#include <cmath>

#define NPTS   320000
#define BATCH  8
#define C_IN   64
#define C_MID  16
#define C_OUT  64
#define H_DS   96
#define W_DS   320
#define HW     (H_DS * W_DS)
#define OUT_ELEMS (BATCH * C_OUT * HW)
#define WAVES_PER_BLOCK 8
#define PTS_PER_TILE 16

typedef float v2f __attribute__((ext_vector_type(2)));
typedef float v8f __attribute__((ext_vector_type(8)));

// ---------------- init output with relu(zero_encoding[c]) ----------------
__global__ void k_init_out(float* __restrict__ out, const float* __restrict__ zero_enc) {
    int i = blockIdx.x * blockDim.x + threadIdx.x;
    if (i >= OUT_ELEMS) return;
    int c = (i / HW) % C_OUT;
    float z = zero_enc[c];
    out[i] = z > 0.f ? z : 0.f;
}

// ---------------- init winner buffer to -1 ----------------
__global__ void k_init_winner(int* __restrict__ winner) {
    int i = blockIdx.x * blockDim.x + threadIdx.x;
    if (i < BATCH * HW) winner[i] = -1;
}

// ---------------- pass 1: project each point, atomicMax winner ----------------
__device__ __forceinline__ int project_flat(const float* __restrict__ cam,
                                            const float* __restrict__ pos,
                                            int b, int p, int* off0_out) {
    const float* K = cam + b * 9;
    float x = pos[(size_t)p * 3 + 0];
    float y = pos[(size_t)p * 3 + 1];
    float zc = pos[(size_t)p * 3 + 2];
    float px = K[0] * x + K[1] * y + K[2] * zc;
    float py = K[3] * x + K[4] * y + K[5] * zc;
    float pz = K[6] * x + K[7] * y + K[8] * zc;
    if (fabsf(pz) < 1e-6f) pz = 1e-6f;
    float uf = fminf(fmaxf(floorf(px / pz), 0.f), (float)(W_DS - 1));
    float vf = fminf(fmaxf(floorf(py / pz), 0.f), (float)(H_DS - 1));
    int u = (int)uf, v = (int)vf;
    *off0_out = b * (C_OUT * HW) + v * W_DS + u;   // NCHW base for c=0
    return b * HW + v * W_DS + u;                  // flat pixel id
}

__global__ void k_project(const float* __restrict__ cam,
                          const float* __restrict__ pos,
                          const int* __restrict__ pbatch,
                          int* __restrict__ winner) {
    int p = blockIdx.x * blockDim.x + threadIdx.x;
    if (p >= NPTS) return;
    int off0;
    int flat = project_flat(cam, pos, pbatch[p], p, &off0);
    atomicMax(&winner[flat], p);
}

// ---------------- pass 2: fused MLP (WMMA f32) + winner scatter ----------------
__global__ __launch_bounds__(WAVES_PER_BLOCK * 32)
void k_mlp_scatter(const float* __restrict__ cam,
                   const float* __restrict__ feats_in,
                   const float* __restrict__ pos,
                   const int* __restrict__ pbatch,
                   const float* __restrict__ w1, const float* __restrict__ b1,
                   const float* __restrict__ w2, const float* __restrict__ b2,
                   const int* __restrict__ winner,
                   float* __restrict__ out) {
    __shared__ float lds_mid[WAVES_PER_BLOCK][PTS_PER_TILE * C_MID]; // [pt][midch]
    __shared__ int   lds_off[WAVES_PER_BLOCK][PTS_PER_TILE];         // NCHW base or -1

    const int wid  = threadIdx.x >> 5;
    const int lane = threadIdx.x & 31;
    const int half = lane >> 4;      // 0: lanes 0-15, 1: lanes 16-31
    const int l16  = lane & 15;
    const int base = (blockIdx.x * WAVES_PER_BLOCK + wid) * PTS_PER_TILE;

    // ---- per-point projection + winner check (all lanes compute, half 0 stores) ----
    {
        int p = base + l16;
        int off0;
        int flat = project_flat(cam, pos, pbatch[p], p, &off0);
        int win  = (winner[flat] == p) ? off0 : -1;
        if (half == 0) lds_off[wid][l16] = win;
    }

    // ---- layer 1: (16 x 64) x (64 x 16), 16 WMMAs with K=4 ----
    // A layout (16x4 f32): VGPR j, lanes0-15 -> row=lane,K=j ; lanes16-31 -> K=j+2
    v8f acc = {0.f, 0.f, 0.f, 0.f, 0.f, 0.f, 0.f, 0.f};
    const float* arow = feats_in + (size_t)(base + l16) * C_IN;
#pragma unroll
    for (int kk = 0; kk < 16; ++kk) {
        int krow = 4 * kk + 2 * half;
        v2f a;  a.x  = arow[krow + 0];           a.y  = arow[krow + 1];
        v2f bm; bm.x = w1[(krow + 0) * C_MID + l16];
                bm.y = w1[(krow + 1) * C_MID + l16];
        acc = __builtin_amdgcn_wmma_f32_16x16x4_f32(
            false, a, false, bm, (short)0, acc, false, false);
    }
    // C layout: VGPR r -> point (r + 8*half), mid channel = l16. bias + relu -> LDS.
    {
        float bias1 = b1[l16];
#pragma unroll
        for (int r = 0; r < 8; ++r) {
            float m = acc[r] + bias1;
            m = m > 0.f ? m : 0.f;
            lds_mid[wid][(r + 8 * half) * C_MID + l16] = m;
        }
    }
    __syncthreads();

    // ---- layer 2: (16 x 16) x (16 x 64), 4 N-chunks x 4 K-chunks ----
    v2f a2[4];
#pragma unroll
    for (int kk2 = 0; kk2 < 4; ++kk2) {
        int kb = 4 * kk2 + 2 * half;
        a2[kk2].x = lds_mid[wid][l16 * C_MID + kb + 0];
        a2[kk2].y = lds_mid[wid][l16 * C_MID + kb + 1];
    }
#pragma unroll
    for (int nc = 0; nc < 4; ++nc) {
        v8f acc2 = {0.f, 0.f, 0.f, 0.f, 0.f, 0.f, 0.f, 0.f};
#pragma unroll
        for (int kk2 = 0; kk2 < 4; ++kk2) {
            int krow = 4 * kk2 + 2 * half;
            v2f bm2; bm2.x = w2[(krow + 0) * C_OUT + nc * 16 + l16];
                     bm2.y = w2[(krow + 1) * C_OUT + nc * 16 + l16];
            acc2 = __builtin_amdgcn_wmma_f32_16x16x4_f32(
                false, a2[kk2], false, bm2, (short)0, acc2, false, false);
        }
        int   c     = nc * 16 + l16;
        float bias2 = b2[c];
#pragma unroll
        for (int r = 0; r < 8; ++r) {
            int q   = r + 8 * half;         // point within tile for VGPR r
            int off = lds_off[wid][q];
            if (off >= 0) {
                float v = acc2[r] + bias2;
                v = v > 0.f ? v : 0.f;
                out[(size_t)off + (size_t)c * HW] = v;
            }
        }
    }
}

extern "C" void kernel_launch(void* const* d_in, const int* in_sizes, int n_in,
                              void* d_out, int out_size, void* d_ws, size_t ws_size,
                              hipStream_t stream) {
    const float* cam      = (const float*)d_in[0];
    const float* pc_feats = (const float*)d_in[1];
    const float* pc_pos   = (const float*)d_in[2];
    const int*   pc_batch = (const int*)d_in[3];
    // d_in[4..6] = B, H, W scalars (compile-time constants here)
    const float* w1       = (const float*)d_in[7];
    const float* b1       = (const float*)d_in[8];
    const float* w2       = (const float*)d_in[9];
    const float* b2       = (const float*)d_in[10];
    const float* zero_enc = (const float*)d_in[11];

    float* out   = (float*)d_out;
    int*   winner = (int*)d_ws;   // BATCH*HW ints = 983 KB

    // 1) fill output with relu(zero_encoding)
    k_init_out<<<(OUT_ELEMS + 255) / 256, 256, 0, stream>>>(out, zero_enc);
    // 2) winner = -1
    k_init_winner<<<(BATCH * HW + 255) / 256, 256, 0, stream>>>(winner);
    // 3) project + atomicMax winner
    k_project<<<(NPTS + 255) / 256, 256, 0, stream>>>(cam, pc_pos, pc_batch, winner);
    // 4) fused WMMA MLP + scatter (16 pts/wave, 8 waves/block; 320000/128 = 2500 blocks)
    k_mlp_scatter<<<NPTS / (WAVES_PER_BLOCK * PTS_PER_TILE), WAVES_PER_BLOCK * 32, 0, stream>>>(
        cam, pc_feats, pc_pos, pc_batch, w1, b1, w2, b2, winner, out);
}